// MHCBlock4_40836549050439
// MI455X (gfx1250) — compile-verified
//
#include <hip/hip_runtime.h>
#include <hip/hip_bf16.h>
#include <math.h>

// ---- problem constants (from reference) ----
#define M_TOTAL 8192
#define K_TOTAL 8192        // S * DIM
#define N_TOTAL 32
#define S_DIM   4
#define DSEG    2048        // DIM
#define SINKHORN_ITERS 20

// ---- tiling ----
#define ROWS    16          // M-tile (WMMA M)
#define KC      256         // K chunk staged in LDS
#define NCHUNK  (K_TOTAL / KC)      // 32
#define THREADS 256         // 8 wave32
#define LDSROW  260         // padded row stride (floats) -> conflict-free A reads

typedef __attribute__((ext_vector_type(2))) float v2f;
typedef __attribute__((ext_vector_type(4))) float f4;
typedef __attribute__((ext_vector_type(8))) float v8f;

__global__ __launch_bounds__(THREADS)
void mhc_fused_kernel(const float* __restrict__ X,
                      const float* __restrict__ W,
                      const float* __restrict__ ab,
                      float* __restrict__ O)
{
    __shared__ float Xs[2][ROWS][LDSROW];      // ~33 KB double-buffered X chunk
    __shared__ float Hbuf[ROWS][N_TOTAL];      // 2 KB  H tile (reduced)
    __shared__ float coefR[ROWS][16];          // sinkhorn result per row
    __shared__ float coefPre[ROWS][4];
    __shared__ float coefPos[ROWS][4];

    const int tid  = threadIdx.x;
    const int lane = tid & 31;
    const int wave = tid >> 5;
    const int row0 = blockIdx.x * ROWS;

    // zero H accumulator
    for (int i = tid; i < ROWS * N_TOTAL; i += THREADS)
        ((float*)Hbuf)[i] = 0.f;

    // ---------------- Stage A: H = X_tile @ W  (WMMA f32 16x16x4) -------------
    const int ntile = wave & 1;          // which 16-wide N tile
    const int kw    = wave >> 1;         // 0..3: k-step phase within chunk
    // A fragment mapping (ISA: lanes0-15 = M rows; VGPR0 holds K=0/K=2 halves)
    const int arow = lane & 15;
    const int akk  = (lane >> 4) << 1;   // 0 or 2
    // B fragment mapping (mirror layout: lane = N col, lane-half = K pair)
    const int bn   = (lane & 15) + ntile * 16;

    float4 stage[4];
    // preload chunk 0
    {
        const float* src = X + (size_t)row0 * K_TOTAL;
        #pragma unroll
        for (int j = 0; j < 4; ++j) {
            int f = tid + j * THREADS;           // float4 index 0..1023
            int r = f >> 6, c4 = f & 63;
            stage[j] = *(const float4*)(src + (size_t)r * K_TOTAL + c4 * 4);
        }
        #pragma unroll
        for (int j = 0; j < 4; ++j) {
            int f = tid + j * THREADS;
            int r = f >> 6, c4 = f & 63;
            *(float4*)&Xs[0][r][c4 * 4] = stage[j];
        }
    }
    __syncthreads();

    v8f acc = {};
    for (int c = 0; c < NCHUNK; ++c) {
        const int cur = c & 1;
        // prefetch next chunk into registers (overlaps with WMMA below)
        if (c + 1 < NCHUNK) {
            const float* src = X + (size_t)row0 * K_TOTAL + (c + 1) * KC;
            #pragma unroll
            for (int j = 0; j < 4; ++j) {
                int f = tid + j * THREADS;
                int r = f >> 6, c4 = f & 63;
                stage[j] = *(const float4*)(src + (size_t)r * K_TOTAL + c4 * 4);
            }
        }
        // 16 k-steps of this wave within the chunk (stride-4 across 4 waves)
        const int kb = c * KC;
        #pragma unroll
        for (int j = 0; j < 16; ++j) {
            const int ks = kw + (j << 2);              // 0..63
            v2f a = *(const v2f*)&Xs[cur][arow][ks * 4 + akk];
            const int kg = kb + ks * 4 + akk;
            v2f b;
            b.x = W[(size_t)kg * N_TOTAL + bn];
            b.y = W[(size_t)(kg + 1) * N_TOTAL + bn];
            acc = __builtin_amdgcn_wmma_f32_16x16x4_f32(
                      false, a, false, b, (short)0, acc, false, false);
        }
        // commit prefetched chunk to the other buffer
        if (c + 1 < NCHUNK) {
            #pragma unroll
            for (int j = 0; j < 4; ++j) {
                int f = tid + j * THREADS;
                int r = f >> 6, c4 = f & 63;
                *(float4*)&Xs[cur ^ 1][r][c4 * 4] = stage[j];
            }
        }
        __syncthreads();
    }

    // reduce 4 partial accumulators per N-tile into Hbuf via LDS float atomics
    {
        const int mhi = (lane >> 4) * 8;    // C layout: VGPR r -> M=r (lo half) / r+8 (hi half)
        #pragma unroll
        for (int r = 0; r < 8; ++r)
            atomicAdd(&Hbuf[r + mhi][bn], acc[r]);
    }
    __syncthreads();

    // ---------------- Stage B: sigmoid gates + Sinkhorn (1 thread / row) ------
    if (tid < ROWS) {
        const int m = tid;
        const float a_res = ab[24], a_pre = ab[25], a_pos = ab[26];
        float P[4][4];
        #pragma unroll
        for (int s = 0; s < 4; ++s)
            #pragma unroll
            for (int i = 0; i < 4; ++i)
                P[s][i] = __expf(a_res * Hbuf[m][s * 4 + i] + ab[s * 4 + i]);
        for (int it = 0; it < SINKHORN_ITERS; ++it) {
            #pragma unroll
            for (int s = 0; s < 4; ++s) {
                float inv = 1.f / (P[s][0] + P[s][1] + P[s][2] + P[s][3]);
                #pragma unroll
                for (int i = 0; i < 4; ++i) P[s][i] *= inv;
            }
            #pragma unroll
            for (int i = 0; i < 4; ++i) {
                float inv = 1.f / (P[0][i] + P[1][i] + P[2][i] + P[3][i]);
                #pragma unroll
                for (int s = 0; s < 4; ++s) P[s][i] *= inv;
            }
        }
        #pragma unroll
        for (int s = 0; s < 4; ++s)
            #pragma unroll
            for (int i = 0; i < 4; ++i)
                coefR[m][s * 4 + i] = P[s][i];
        #pragma unroll
        for (int i = 0; i < 4; ++i) {
            float hp = Hbuf[m][16 + i];
            coefPre[m][i] = a_pre * (1.f / (1.f + __expf(-hp))) + ab[16 + i];
            float hq = Hbuf[m][20 + i];
            coefPos[m][i] = a_pos * (1.f / (1.f + __expf(-hq))) + ab[20 + i];
        }
    }
    __syncthreads();

    // ---------------- Stage C: mixing pass (streaming, float4 coalesced) ------
    const float* Xb = X + (size_t)row0 * K_TOTAL;
    float*       Ob = O + (size_t)row0 * K_TOTAL;
    const int NITER = (ROWS * DSEG / 4) / THREADS;   // 32
    for (int it = 0; it < NITER; ++it) {
        const int idx = it * THREADS + tid;          // float4 task id
        const int m  = idx >> 9;                     // / (DSEG/4)
        const int dv = idx & 511;
        const float* xp = Xb + (size_t)m * K_TOTAL + dv * 4;
        f4 x0 = *(const f4*)(xp);
        f4 x1 = *(const f4*)(xp + DSEG);
        f4 x2 = *(const f4*)(xp + 2 * DSEG);
        f4 x3 = *(const f4*)(xp + 3 * DSEG);
        const float p0 = coefPre[m][0], p1 = coefPre[m][1];
        const float p2 = coefPre[m][2], p3 = coefPre[m][3];
        f4 y = p0 * x0 + p1 * x1 + p2 * x2 + p3 * x3;
        float* op = Ob + (size_t)m * K_TOTAL + dv * 4;
        #pragma unroll
        for (int s = 0; s < 4; ++s) {
            const float r0 = coefR[m][s * 4 + 0], r1 = coefR[m][s * 4 + 1];
            const float r2 = coefR[m][s * 4 + 2], r3 = coefR[m][s * 4 + 3];
            f4 o = r0 * x0 + r1 * x1 + r2 * x2 + r3 * x3 + coefPos[m][s] * y;
            *(f4*)(op + (size_t)s * DSEG) = o;
        }
    }
}

extern "C" void kernel_launch(void* const* d_in, const int* in_sizes, int n_in,
                              void* d_out, int out_size, void* d_ws, size_t ws_size,
                              hipStream_t stream) {
    (void)in_sizes; (void)n_in; (void)out_size; (void)d_ws; (void)ws_size;
    const float* X  = (const float*)d_in[0];
    const float* W  = (const float*)d_in[1];
    const float* ab = (const float*)d_in[2];
    float* O = (float*)d_out;
    dim3 grid(M_TOTAL / ROWS);   // 512 workgroups
    mhc_fused_kernel<<<grid, THREADS, 0, stream>>>(X, W, ab, O);
}